// LinearLoss_34711925686323
// MI455X (gfx1250) — compile-verified
//
#include <hip/hip_runtime.h>
#include <hip/hip_bf16.h>

// Shapes from the reference
#define R0 8192     // theta0 rows
#define C0 4096     // theta0 cols
#define R1 4096     // theta1 rows
#define C1 2048     // theta1 cols
#define NG 1024     // segment groups
#define NM 512      // mapping1 rows / obs1 len

#define COLS 32     // theta0 columns per block (LDS bins = 1024*32*4 = 128 KB)

typedef float v2f __attribute__((ext_vector_type(2)));
typedef float v8f __attribute__((ext_vector_type(8)));

// -------- K0: zero the three loss accumulators --------
__global__ void k_zero(float* __restrict__ acc) {
    if (threadIdx.x < 3) acc[threadIdx.x] = 0.0f;
}

// -------- K1: fused theta0 pass: exp + LDS segment scatter + loss_a + loss_c ----
// grid = C0/COLS = 128 blocks, block = (32, 8) = 256 threads (8 waves)
__global__ __launch_bounds__(256) void k_theta0(const float* __restrict__ th0,
                                                const float* __restrict__ obs0,
                                                const float* __restrict__ obs2,
                                                const int*   __restrict__ idx0,
                                                float* __restrict__ acc) {
    __shared__ float bins[NG * COLS];   // 128 KB: all 1024 groups x 32 cols
    __shared__ float red[256];

    const int tx      = threadIdx.x;            // 0..31 (column within tile)
    const int ty      = threadIdx.y;            // 0..7  (wave id)
    const int flat    = ty * 32 + tx;
    const int colbase = blockIdx.x * COLS;
    const int col     = colbase + tx;

    for (int i = flat; i < NG * COLS; i += 256) bins[i] = 0.0f;
    __syncthreads();

    // Stream all rows for our 32 columns: exp -> LDS atomic scatter + column sum
    float csum = 0.0f;
    for (int r = ty; r < R0; r += 8) {
        if (r + 64 < R0)
            __builtin_prefetch(&th0[(size_t)(r + 64) * C0 + col], 0, 1);
        float e = __expf(th0[(size_t)r * C0 + col]);
        int   g = idx0[r];                       // uniform across the wave
        atomicAdd(&bins[g * COLS + tx], e);      // ds_add_f32
        csum += e;
    }

    red[flat] = csum;
    __syncthreads();                             // bins + red now stable

    // ---- loss_c partial: column sums live in red[ty][tx], wave 0 reduces ----
    if (ty == 0) {
        float s = 0.0f;
        #pragma unroll
        for (int j = 0; j < 8; ++j) s += red[j * 32 + tx];
        float d  = obs2[col] - s;
        float lc = d * d;
        #pragma unroll
        for (int o = 16; o > 0; o >>= 1) lc += __shfl_xor(lc, o);
        if (tx == 0) atomicAdd(&acc[2], lc);
    }

    // ---- loss_a partial: (obs0 - bins)^2 over 1024 x 32 tile ----
    float la = 0.0f;
    for (int i = flat; i < NG * COLS; i += 256) {
        int   g = i / COLS;
        int   c = i - g * COLS;
        float d = obs0[(size_t)g * C0 + colbase + c] - bins[i];
        la += d * d;
    }
    #pragma unroll
    for (int o = 16; o > 0; o >>= 1) la += __shfl_xor(la, o);
    if (tx == 0) atomicAdd(&acc[0], la);
}

// -------- K2: p1[r] = sum_c exp(theta1[r,c]) ; one wave per row --------
// grid = R1/8 = 512 blocks, block = (32, 8)
__global__ __launch_bounds__(256) void k_theta1(const float* __restrict__ th1,
                                                float* __restrict__ p1) {
    const int row  = blockIdx.x * 8 + threadIdx.y;
    const int lane = threadIdx.x;
    float s = 0.0f;
    for (int c = lane; c < C1; c += 32)
        s += __expf(th1[(size_t)row * C1 + c]);
    #pragma unroll
    for (int o = 16; o > 0; o >>= 1) s += __shfl_xor(s, o);
    if (lane == 0) p1[row] = s;
}

// -------- K3: m1 = mapping1 @ p1 via V_WMMA_F32_16X16X4_F32, + loss_b --------
// grid = NM/16 = 32 blocks, block = 32 threads (one wave, EXEC all ones)
// A layout (ISA 7.12.2, 32-bit A 16x4): lane L -> M = L%16, vgpr v -> K = v + (L<16?0:2)
// B layout (4x16, row striped across lanes): vgpr v, lanes<16 -> K=v, lanes>=16 -> K=v+2.
// We replicate p1 across all N columns, so every column of D equals the GEMV result.
__global__ __launch_bounds__(32) void k_gemv(const float* __restrict__ map,
                                             const float* __restrict__ p1,
                                             const float* __restrict__ obs1,
                                             float* __restrict__ acc) {
    const int lane    = threadIdx.x;
    const int rowbase = blockIdx.x * 16;
    const int m       = lane & 15;
    const int kh      = (lane < 16) ? 0 : 2;
    const float* __restrict__ arow = map + (size_t)(rowbase + m) * C0;

    v8f c = {};
    for (int k = 0; k < C0; k += 4) {
        v2f a, b;
        a.x = arow[k + kh];
        a.y = arow[k + kh + 1];
        b.x = p1[k + kh];
        b.y = p1[k + kh + 1];
        c = __builtin_amdgcn_wmma_f32_16x16x4_f32(
                /*neg_a=*/false, a, /*neg_b=*/false, b,
                /*c_mod=*/(short)0, c, /*reuse_a=*/false, /*reuse_b=*/false);
    }

    // D: lane L, vgpr v -> M = v + (L<16?0:8), N = L%16. Take N==0 lanes (0 and 16).
    if ((lane & 15) == 0) {
        const int mbase = rowbase + ((lane < 16) ? 0 : 8);
        float lb = 0.0f;
        #pragma unroll
        for (int v = 0; v < 8; ++v) {
            float d = obs1[mbase + v] - c[v];
            lb += d * d;
        }
        atomicAdd(&acc[1], lb);
    }
}

// -------- K4: final scalar --------
__global__ void k_final(const float* __restrict__ acc, float* __restrict__ out) {
    if (threadIdx.x == 0) {
        float la = acc[0] * (1.0f / ((float)NG * (float)C0));
        float lb = acc[1] * (1.0f / (float)NM);
        float lc = 0.5f * acc[2] * (1.0f / (float)C0);
        out[0] = (la + lb + lc) * (1.0f / 3.0f);
    }
}

extern "C" void kernel_launch(void* const* d_in, const int* in_sizes, int n_in,
                              void* d_out, int out_size, void* d_ws, size_t ws_size,
                              hipStream_t stream) {
    const float* theta0   = (const float*)d_in[0];
    const float* theta1   = (const float*)d_in[1];
    const float* obs0     = (const float*)d_in[2];
    const float* obs1     = (const float*)d_in[3];
    const float* obs2     = (const float*)d_in[4];
    const int*   idx0     = (const int*)  d_in[5];
    const float* mapping1 = (const float*)d_in[6];
    float* out = (float*)d_out;

    float* acc = (float*)d_ws;                       // 3 loss accumulators
    float* p1  = (float*)((char*)d_ws + 256);        // [4096] row sums of exp(theta1)

    k_zero  <<<1, 32, 0, stream>>>(acc);
    k_theta0<<<dim3(C0 / COLS), dim3(32, 8), 0, stream>>>(theta0, obs0, obs2, idx0, acc);
    k_theta1<<<dim3(R1 / 8),    dim3(32, 8), 0, stream>>>(theta1, p1);
    k_gemv  <<<dim3(NM / 16),   dim3(32),    0, stream>>>(mapping1, p1, obs1, acc);
    k_final <<<1, 32, 0, stream>>>(acc, out);
}